// GraphSMOTE_90391881711713
// MI455X (gfx1250) — compile-verified
//
#include <hip/hip_runtime.h>

// Problem sizes (match reference)
#define NN_  8192   // nodes
#define DD_  256    // embedding dim
#define FF_  1536   // fraud count
#define GG_  1536   // generated count
#define HH_  512    // hidden dim (2*D)
#define TT_  9728   // N + G

typedef __attribute__((ext_vector_type(2))) float v2f;
typedef __attribute__((ext_vector_type(4))) float v4f;   // native vector: OK for NT builtins
typedef __attribute__((ext_vector_type(8))) float v8f;

// Exact-fp32 WMMA: D(16x16,f32) = A(16x4,f32) * B(4x16,f32) + C
__device__ __forceinline__ v8f wmma_f32x4(v2f a, v2f b, v8f c) {
  return __builtin_amdgcn_wmma_f32_16x16x4_f32(false, a, false, b, (short)0, c,
                                               false, false);
}

// ---------------------------------------------------------------------------
// Memory-bound assembly kernels (~650MB HBM traffic -> these set the runtime).
// All one-shot streams are non-temporal so L2 keeps emb/T hot for the GEMMs.
// ---------------------------------------------------------------------------
__global__ __launch_bounds__(256) void k_copy_emb(const v4f* __restrict__ in,
                                                  v4f* __restrict__ out) {
  int i = blockIdx.x * blockDim.x + threadIdx.x;     // N*D/4 = 524288
  v4f v = in[i];                        // regular load: warms L2 for k_edges
  __builtin_nontemporal_store(v, &out[i]);
}

__global__ __launch_bounds__(256) void k_labels(const int* __restrict__ labels,
                                                float* __restrict__ outLab) {
  int i = blockIdx.x * blockDim.x + threadIdx.x;
  if (i < TT_) outLab[i] = (i < NN_) ? (float)labels[i] : 1.0f;
}

__global__ __launch_bounds__(256) void k_copy_adj(const v4f* __restrict__ adj4,
                                                  v4f* __restrict__ outAdj4) {
  int i = blockIdx.x * blockDim.x + threadIdx.x;     // N*N/4 = 16777216
  int row = i >> 11;            // / (N/4)
  int c4  = i & 2047;           // % (N/4)
  v4f v = __builtin_nontemporal_load(&adj4[(size_t)row * (NN_ / 4) + c4]);
  __builtin_nontemporal_store(v, &outAdj4[(size_t)row * (TT_ / 4) + c4]);
}

__global__ __launch_bounds__(256) void k_zero_corner(v4f* __restrict__ outAdj4) {
  int i = blockIdx.x * blockDim.x + threadIdx.x;     // G*G/4 = 589824
  int r  = i / (GG_ / 4);
  int c4 = i % (GG_ / 4);
  v4f z = {0.f, 0.f, 0.f, 0.f};
  __builtin_nontemporal_store(z, &outAdj4[(size_t)(NN_ + r) * (TT_ / 4) + (NN_ / 4) + c4]);
}

// ---------------------------------------------------------------------------
// Fraud norms: ||fraud[j]||^2
// ---------------------------------------------------------------------------
__global__ __launch_bounds__(256) void k_norms(const float* __restrict__ emb,
                                               const int* __restrict__ fraud_idx,
                                               float* __restrict__ norms) {
  int j = blockIdx.x * blockDim.x + threadIdx.x;
  if (j >= FF_) return;
  const float4* r = (const float4*)(emb + (size_t)fraud_idx[j] * DD_);
  float s = 0.f;
#pragma unroll 8
  for (int k = 0; k < DD_ / 4; ++k) {
    float4 v = r[k];
    s += v.x * v.x + v.y * v.y + v.z * v.z + v.w * v.w;
  }
  norms[j] = s;
}

// ---------------------------------------------------------------------------
// Second-nearest-neighbor search with top_k-compatible (value,index) ordering
// ---------------------------------------------------------------------------
__global__ __launch_bounds__(256) void k_nn(const float* __restrict__ emb,
                                            const int* __restrict__ fraud_idx,
                                            const int* __restrict__ base_idx,
                                            const float* __restrict__ norms,
                                            int* __restrict__ nn_out) {
  __shared__ __align__(16) float sbase[DD_];
  __shared__ float rv1[256], rv2[256];
  __shared__ int   ri1[256], ri2[256];
  int g = blockIdx.x, tid = threadIdx.x;
  int bi = base_idx[g];
  const float* brow = emb + (size_t)fraud_idx[bi] * DD_;
  sbase[tid] = brow[tid];           // blockDim == D == 256
  __syncthreads();
  float nb = norms[bi];
  float v1 = 3.4e38f, v2 = 3.4e38f;
  int i1 = 0x7fffffff, i2 = 0x7fffffff;
  const float4* bs = (const float4*)sbase;
  for (int j = tid; j < FF_; j += 256) {
    const float4* fr = (const float4*)(emb + (size_t)fraud_idx[j] * DD_);
    float dot = 0.f;
#pragma unroll 8
    for (int k = 0; k < DD_ / 4; ++k) {
      float4 f = fr[k], b = bs[k];
      dot += f.x * b.x + f.y * b.y + f.z * b.z + f.w * b.w;
    }
    float d2 = nb + norms[j] - 2.0f * dot;
    if (d2 < v1 || (d2 == v1 && j < i1)) { v2 = v1; i2 = i1; v1 = d2; i1 = j; }
    else if (d2 < v2 || (d2 == v2 && j < i2)) { v2 = d2; i2 = j; }
  }
  rv1[tid] = v1; ri1[tid] = i1; rv2[tid] = v2; ri2[tid] = i2;
  __syncthreads();
  for (int s = 128; s > 0; s >>= 1) {
    if (tid < s) {
      float w1 = rv1[tid + s]; int j1 = ri1[tid + s];
      float w2 = rv2[tid + s]; int j2 = ri2[tid + s];
      float a1 = rv1[tid]; int a1i = ri1[tid];
      float a2 = rv2[tid]; int a2i = ri2[tid];
      if (w1 < a1 || (w1 == a1 && j1 < a1i)) { a2 = a1; a2i = a1i; a1 = w1; a1i = j1; }
      else if (w1 < a2 || (w1 == a2 && j1 < a2i)) { a2 = w1; a2i = j1; }
      if (w2 < a2 || (w2 == a2 && j2 < a2i)) { a2 = w2; a2i = j2; }
      rv1[tid] = a1; ri1[tid] = a1i; rv2[tid] = a2; ri2[tid] = a2i;
    }
    __syncthreads();
  }
  if (tid == 0) nn_out[g] = ri2[0];   // nn2[:,1]
}

// ---------------------------------------------------------------------------
// interp = (1-d)*base + d*neighbor
// ---------------------------------------------------------------------------
__global__ __launch_bounds__(256) void k_interp(const float* __restrict__ emb,
                                                const int* __restrict__ fraud_idx,
                                                const int* __restrict__ base_idx,
                                                const int* __restrict__ nn,
                                                const float* __restrict__ deltas,
                                                float4* __restrict__ interp4) {
  int i = blockIdx.x * blockDim.x + threadIdx.x;     // G*D/4 = 98304
  if (i >= GG_ * DD_ / 4) return;
  int g  = i / (DD_ / 4);
  int k4 = i % (DD_ / 4);
  float d = deltas[g];
  const float4* b = (const float4*)(emb + (size_t)fraud_idx[base_idx[g]] * DD_);
  const float4* n = (const float4*)(emb + (size_t)fraud_idx[nn[g]] * DD_);
  float4 bb = b[k4], nv = n[k4], o;
  float od = 1.0f - d;
  o.x = od * bb.x + d * nv.x; o.y = od * bb.y + d * nv.y;
  o.z = od * bb.z + d * nv.z; o.w = od * bb.w + d * nv.w;
  interp4[i] = o;
}

// ---------------------------------------------------------------------------
// fp32-WMMA GEMM, compile-time shapes: out = act(A[M,K] @ B[K,Nc] + bias)
// 8 waves/block; each wave owns TWO 16x16 tiles (shared A, two acc chains),
// so a block covers 16 rows x 256 cols with immediate-offset addressing.
// ---------------------------------------------------------------------------
template <int K, int Nc, bool RELU, bool BIAS, bool OUT2>
__global__ __launch_bounds__(256) void k_gemm_wmma(const float* __restrict__ A,
                                                   const float* __restrict__ B,
                                                   const float* __restrict__ bias,
                                                   float* __restrict__ out,
                                                   float* __restrict__ out2) {
  int tid = threadIdx.x, wave = tid >> 5, lane = tid & 31;
  int m0 = blockIdx.x * 16;
  int n0 = blockIdx.y * 256 + wave * 32;
  int half = lane >> 4, l16 = lane & 15, kk = half * 2;
  const float* arow = A + (size_t)(m0 + l16) * K;
  const float* bcol = B + (size_t)kk * Nc + n0 + l16;   // column n, row kk
  v8f acc0 = {}, acc1 = {};
#pragma unroll 16
  for (int k0 = 0; k0 < K; k0 += 4) {
    v2f a = *(const v2f*)(arow + k0 + kk);              // A[m][k0+kk .. +1]
    v2f b0, b1;
    b0.x = bcol[(size_t)k0 * Nc];                       // B[k0+kk  ][n0+l16]
    b0.y = bcol[(size_t)(k0 + 1) * Nc];                 // B[k0+kk+1][n0+l16]
    b1.x = bcol[(size_t)k0 * Nc + 16];
    b1.y = bcol[(size_t)(k0 + 1) * Nc + 16];
    acc0 = wmma_f32x4(a, b0, acc0);
    acc1 = wmma_f32x4(a, b1, acc1);
  }
  int col0 = n0 + l16;
  float bv0 = BIAS ? bias[col0] : 0.0f;
  float bv1 = BIAS ? bias[col0 + 16] : 0.0f;
#pragma unroll
  for (int r = 0; r < 8; ++r) {
    int row = m0 + half * 8 + r;
    float v0 = acc0[r] + bv0, v1 = acc1[r] + bv1;
    if (RELU) { v0 = fmaxf(v0, 0.0f); v1 = fmaxf(v1, 0.0f); }
    out[(size_t)row * Nc + col0]      = v0;
    out[(size_t)row * Nc + col0 + 16] = v1;
    if (OUT2) {   // all_emb copy: never re-read -> stream it
      __builtin_nontemporal_store(v0, &out2[(size_t)row * Nc + col0]);
      __builtin_nontemporal_store(v1, &out2[(size_t)row * Nc + col0 + 16]);
    }
  }
}

// ---------------------------------------------------------------------------
// edges = (T[G,D] @ emb.T > 0); writes new_adj[N+g, n] and new_adj[n, N+g]
// Per wave: two 16x16 logit tiles (shared A). Transposed half goes through a
// padded LDS tile so both store patterns are 64B-contiguous per half-wave.
// ---------------------------------------------------------------------------
__global__ __launch_bounds__(256) void k_edges(const float* __restrict__ T,
                                               const float* __restrict__ emb,
                                               float* __restrict__ outAdj) {
  __shared__ float tiles[8][2][16][17];
  int tid = threadIdx.x, wave = tid >> 5, lane = tid & 31;
  int m0 = blockIdx.x * 16;              // 96 tiles over G
  int n0 = blockIdx.y * 256 + wave * 32; // 32 chunks over N
  int half = lane >> 4, l16 = lane & 15, kk = half * 2;
  const float* arow  = T   + (size_t)(m0 + l16) * DD_;
  const float* brow0 = emb + (size_t)(n0 + l16) * DD_;       // B[k][n]=emb[n][k]
  const float* brow1 = brow0 + 16 * DD_;
  v8f acc0 = {}, acc1 = {};
#pragma unroll 16
  for (int k0 = 0; k0 < DD_; k0 += 4) {
    v2f a  = *(const v2f*)(arow  + k0 + kk);
    v2f b0 = *(const v2f*)(brow0 + k0 + kk);
    v2f b1 = *(const v2f*)(brow1 + k0 + kk);
    acc0 = wmma_f32x4(a, b0, acc0);
    acc1 = wmma_f32x4(a, b1, acc1);
  }
#pragma unroll
  for (int r = 0; r < 8; ++r) {
    int mr = half * 8 + r;
    float e0 = (acc0[r] > 0.0f) ? 1.0f : 0.0f;   // sigmoid(x)>0.5 <=> x>0
    float e1 = (acc1[r] > 0.0f) ? 1.0f : 0.0f;
    float* rowp = outAdj + (size_t)(NN_ + m0 + mr) * TT_ + n0;
    __builtin_nontemporal_store(e0, rowp + l16);
    __builtin_nontemporal_store(e1, rowp + 16 + l16);
    tiles[wave][0][mr][l16] = e0;
    tiles[wave][1][mr][l16] = e1;
  }
  __syncthreads();
#pragma unroll
  for (int t = 0; t < 2; ++t) {
#pragma unroll
    for (int i = 0; i < 8; ++i) {
      int idx = i * 32 + lane;
      int nl = idx >> 4, ml = idx & 15;
      __builtin_nontemporal_store(tiles[wave][t][ml][nl],
          &outAdj[(size_t)(n0 + t * 16 + nl) * TT_ + (NN_ + m0 + ml)]);
    }
  }
}

// ---------------------------------------------------------------------------
extern "C" void kernel_launch(void* const* d_in, const int* in_sizes, int n_in,
                              void* d_out, int out_size, void* d_ws, size_t ws_size,
                              hipStream_t stream) {
  const float* emb       = (const float*)d_in[0];
  const int*   labels    = (const int*)  d_in[1];
  const float* adj       = (const float*)d_in[2];
  const int*   fraud_idx = (const int*)  d_in[3];
  const int*   base_idx  = (const int*)  d_in[4];
  const float* deltas    = (const float*)d_in[5];
  const float* W1        = (const float*)d_in[6];
  const float* b1        = (const float*)d_in[7];
  const float* W2        = (const float*)d_in[8];
  const float* b2        = (const float*)d_in[9];
  const float* W_edge    = (const float*)d_in[10];

  float* out    = (float*)d_out;
  float* outEmb = out;
  float* outLab = out + (size_t)TT_ * DD_;
  float* outAdj = outLab + TT_;

  float* w      = (float*)d_ws;
  float* norms  = w;                          // 1536 f
  int*   nnIdx  = (int*)(w + 2048);           // 1536 i
  float* interp = w + 4096;                   // G*D   = 393216 f
  float* hbuf   = interp + (size_t)GG_ * DD_; // G*H   = 786432 f
  float* newEmb = hbuf   + (size_t)GG_ * HH_; // G*D
  float* Tbuf   = newEmb + (size_t)GG_ * DD_; // G*D   (total ~7.9 MB)

  // Assembly / copies (memory-bound critical path, NT-streamed)
  k_copy_emb   <<<(NN_ * DD_ / 4) / 256, 256, 0, stream>>>((const v4f*)emb, (v4f*)outEmb);
  k_labels     <<<(TT_ + 255) / 256, 256, 0, stream>>>(labels, outLab);
  k_copy_adj   <<<(NN_ * NN_ / 4) / 256, 256, 0, stream>>>((const v4f*)adj, (v4f*)outAdj);
  k_zero_corner<<<(GG_ * GG_ / 4) / 256, 256, 0, stream>>>((v4f*)outAdj);

  // SMOTE pipeline
  k_norms <<<(FF_ + 255) / 256, 256, 0, stream>>>(emb, fraud_idx, norms);
  k_nn    <<<GG_, 256, 0, stream>>>(emb, fraud_idx, base_idx, norms, nnIdx);
  k_interp<<<(GG_ * DD_ / 4 + 255) / 256, 256, 0, stream>>>(emb, fraud_idx, base_idx,
                                                            nnIdx, deltas, (float4*)interp);

  // h = relu(interp @ W1 + b1)                 M=1536 K=256 N=512
  k_gemm_wmma<DD_, HH_, true, true, false>
      <<<dim3(GG_ / 16, HH_ / 256), 256, 0, stream>>>(interp, W1, b1, hbuf, nullptr);
  // new_emb = h @ W2 + b2 (also into all_emb)  M=1536 K=512 N=256
  k_gemm_wmma<HH_, DD_, false, true, true>
      <<<dim3(GG_ / 16, 1), 256, 0, stream>>>(hbuf, W2, b2, newEmb,
                                              outEmb + (size_t)NN_ * DD_);
  // T = new_emb @ W_edge                       M=1536 K=256 N=256
  k_gemm_wmma<DD_, DD_, false, false, false>
      <<<dim3(GG_ / 16, 1), 256, 0, stream>>>(newEmb, W_edge, nullptr, Tbuf, nullptr);
  // edges + both adjacency blocks              M=1536 K=256 N=8192
  k_edges<<<dim3(GG_ / 16, NN_ / 256), 256, 0, stream>>>(Tbuf, emb, outAdj);
}